// DistPtsTopo_69020124447083
// MI455X (gfx1250) — compile-verified
//
#include <hip/hip_runtime.h>

// Reference: DistPtsTopo — per point: cell, 8 jittered corners, 12 edge mids,
// 48 triangle centroids (fixed linear map W[48][8] of corner positions),
// squared distance point->centroid, segment-sum by cell into (64^3, 48).
//
// WMMA mapping: cent(48x3 per point) = W(48x8) x corner_pos(8x3).
// Per wave: A = W tile (16 tris x 4 corners per K-step), B = corner positions
// (4 corners x 16 points), C = centroids. f32 WMMA (16x16x4) keeps exact
// reference precision (W entries are multiples of 1/6).

#define NGRID 65
#define NCELL 64
#define NTRI  48
#define G3    (NGRID * NGRID * NGRID)

typedef float v2f __attribute__((ext_vector_type(2)));
typedef float v8f __attribute__((ext_vector_type(8)));

// Corner index c = i*4 + j*2 + k  (i,j,k in {0,1}), matching CORNERS order.
__constant__ int d_EDGES[12][2] = {
    {0,1},{0,2},{0,4},{1,3},{1,5},{2,3},
    {2,6},{3,7},{4,5},{4,6},{5,7},{6,7}};

// First 48 of combinations(range(12), 3) in lexicographic order.
__constant__ int d_TRIS[48][3] = {
    {0,1,2},{0,1,3},{0,1,4},{0,1,5},{0,1,6},{0,1,7},{0,1,8},{0,1,9},{0,1,10},{0,1,11},
    {0,2,3},{0,2,4},{0,2,5},{0,2,6},{0,2,7},{0,2,8},{0,2,9},{0,2,10},{0,2,11},
    {0,3,4},{0,3,5},{0,3,6},{0,3,7},{0,3,8},{0,3,9},{0,3,10},{0,3,11},
    {0,4,5},{0,4,6},{0,4,7},{0,4,8},{0,4,9},{0,4,10},{0,4,11},
    {0,5,6},{0,5,7},{0,5,8},{0,5,9},{0,5,10},{0,5,11},
    {0,6,7},{0,6,8},{0,6,9},{0,6,10},{0,6,11},
    {0,7,8},{0,7,9},{0,7,10}};

// W[t][c] = (#occurrences of corner c in triangle t's three edges) / 6
__device__ __forceinline__ float wval(int t, int c) {
    int cnt = 0;
#pragma unroll
    for (int e = 0; e < 3; ++e) {
        int ed = d_TRIS[t][e];
        cnt += (d_EDGES[ed][0] == c) ? 1 : 0;
        cnt += (d_EDGES[ed][1] == c) ? 1 : 0;
    }
    return (float)cnt * (1.0f / 6.0f);
}

__global__ void zero_out_kernel(float4* __restrict__ out, int n4) {
    int i = blockIdx.x * blockDim.x + threadIdx.x;
    int stride = gridDim.x * blockDim.x;
    for (; i < n4; i += stride) out[i] = make_float4(0.f, 0.f, 0.f, 0.f);
}

__global__ void __launch_bounds__(192)
dist_tris_kernel(const float* __restrict__ off,   // (3,65,65,65)
                 const float* __restrict__ pts,   // (P,3)
                 float* __restrict__ out,         // (64^3, 48)
                 int npts) {
    const int lane = threadIdx.x & 31;
    const int wave = threadIdx.x >> 5;
    const int hi   = lane >> 4;    // half-wave id: selects j-bit / K pair / C row half
    const int pt   = lane & 15;    // point slot (N) == triangle row slot of A (M uses same)

    int gp = (blockIdx.x * (int)(blockDim.x >> 5) + wave) * 16 + pt;
    const bool active = (gp < npts);
    gp = active ? gp : (npts - 1);   // clamp loads; no branch, EXEC stays full

    const float px = pts[gp * 3 + 0];
    const float py = pts[gp * 3 + 1];
    const float pz = pts[gp * 3 + 2];

    const int cx = min(max((int)floorf(px), 0), NCELL - 1);
    const int cy = min(max((int)floorf(py), 0), NCELL - 1);
    const int cz = min(max((int)floorf(pz), 0), NCELL - 1);
    const int cellflat = (cx * NCELL + cy) * NCELL + cz;

    // Gather the 4 corners this lane owns (j = hi): cpv[i][k][comp]
    float cpv[2][2][3];
#pragma unroll
    for (int i = 0; i < 2; ++i) {
#pragma unroll
        for (int k = 0; k < 2; ++k) {
            const int X = cx + i, Y = cy + hi, Z = cz + k;
            const int b3 = (X * NGRID + Y) * NGRID + Z;
            cpv[i][k][0] = (float)X + off[b3];
            cpv[i][k][1] = (float)Y + off[G3 + b3];
            cpv[i][k][2] = (float)Z + off[2 * G3 + b3];
        }
    }

    // B operand per (K-step s, component): corner c = s*4 + hi*2 + r  (r = vgpr slot)
    v2f bmat[2][3];
#pragma unroll
    for (int s = 0; s < 2; ++s) {
#pragma unroll
        for (int comp = 0; comp < 3; ++comp) {
            v2f b;
            b.x = cpv[s][0][comp];
            b.y = cpv[s][1][comp];
            bmat[s][comp] = b;
        }
    }

    const float pcoord[3] = {px, py, pz};

#pragma unroll
    for (int tile = 0; tile < 3; ++tile) {
        const int t = tile * 16 + pt;   // A row (M) = lane&15 per 32-bit A layout
        v2f amat[2];
#pragma unroll
        for (int s = 0; s < 2; ++s) {
            const int cbase = s * 4 + hi * 2;
            v2f a;
            a.x = wval(t, cbase + 0);
            a.y = wval(t, cbase + 1);
            amat[s] = a;
        }

        v8f dacc = {0.f, 0.f, 0.f, 0.f, 0.f, 0.f, 0.f, 0.f};
#pragma unroll
        for (int comp = 0; comp < 3; ++comp) {
            v8f cent = {0.f, 0.f, 0.f, 0.f, 0.f, 0.f, 0.f, 0.f};
            // cent(16 tris x 16 pts) += W_tile x corner_pos, K = 8 in two K=4 steps
            cent = __builtin_amdgcn_wmma_f32_16x16x4_f32(
                false, amat[0], false, bmat[0][comp], (short)0, cent, false, false);
            cent = __builtin_amdgcn_wmma_f32_16x16x4_f32(
                false, amat[1], false, bmat[1][comp], (short)0, cent, false, false);
            const float pc = pcoord[comp];
#pragma unroll
            for (int v = 0; v < 8; ++v) {
                const float df = pc - cent[v];
                dacc[v] += df * df;
            }
        }

        if (active) {
#pragma unroll
            for (int v = 0; v < 8; ++v) {
                const int tri = tile * 16 + hi * 8 + v;  // C row M = v + hi*8
                atomicAdd(out + (size_t)cellflat * NTRI + tri, dacc[v]);
            }
        }
    }
}

extern "C" void kernel_launch(void* const* d_in, const int* in_sizes, int n_in,
                              void* d_out, int out_size, void* d_ws, size_t ws_size,
                              hipStream_t stream) {
    const float* off = (const float*)d_in[0];   // (3,65,65,65) f32
    const float* pts = (const float*)d_in[1];   // (P,3) f32
    float* out = (float*)d_out;                 // (64^3, 48) f32
    const int npts = in_sizes[1] / 3;

    // Zero the accumulator output (stream-ordered, graph-capture safe).
    const int n4 = out_size / 4;   // 12,582,912 % 4 == 0
    zero_out_kernel<<<2048, 256, 0, stream>>>((float4*)out, n4);

    // 6 waves/block (192 threads), 16 points per wave -> 96 points/block.
    // 300000 = 3125 * 96 exactly: every wave is full, EXEC all-1s for WMMA.
    const int wavesPerBlock = 6;
    const int ptsPerBlock = wavesPerBlock * 16;
    const int blocks = (npts + ptsPerBlock - 1) / ptsPerBlock;
    dist_tris_kernel<<<blocks, wavesPerBlock * 32, 0, stream>>>(off, pts, out, npts);
}